// HopfieldAssociativeMemory_5531917877582
// MI455X (gfx1250) — compile-verified
//
#include <hip/hip_runtime.h>
#include <math.h>

// ---------------------------------------------------------------------------
// CDNA5 (gfx1250) f32 WMMA: V_WMMA_F32_16X16X4_F32
//   A: 16x4  -> 2 VGPRs/lane  : a[j] = A[l%16][2*(l/16)+j]
//   B: 4x16  -> 2 VGPRs/lane  : b[j] = B[2*(l/16)+j][l%16]
//   C: 16x16 -> 8 VGPRs/lane  : c[r] = C[r + 8*(l/16)][l%16]
// ---------------------------------------------------------------------------
typedef float v2f __attribute__((ext_vector_type(2)));
typedef float v8f __attribute__((ext_vector_type(8)));

__device__ __forceinline__ v8f wmma_f32(v2f a, v2f b, v8f c) {
  return __builtin_amdgcn_wmma_f32_16x16x4_f32(
      /*neg_a=*/false, a, /*neg_b=*/false, b,
      /*c_mod=*/(short)0, c, /*reuse_a=*/false, /*reuse_b=*/false);
}

#define BATCH      256
#define DIM        512
#define NHEADS     8
#define NSTORED    4096
#define NMEM       100000
#define TOPK       5
#define SCALING    0.8f
#define EPS2       1e-8f      // EPS^2, EPS = 1e-4
#define MAX_ITERS  50
#define NEG_INF    (-3.0e38f)

// ---------------------------------------------------------------------------
// Generic WMMA GEMM: C[M,N] = A[M,K] @ B[K,N] (+bias, optional relu).
// One wave per 16x16 output tile. M,N multiples of 16; K multiple of 4.
// ---------------------------------------------------------------------------
__global__ void gemm_wmma_kernel(const float* __restrict__ A, int lda,
                                 const float* __restrict__ B, int ldb,
                                 float* __restrict__ C, int ldc,
                                 const float* __restrict__ bias, int relu,
                                 int M, int N, int K) {
  int tiles_n = N >> 4;
  int tile    = blockIdx.x * blockDim.y + threadIdx.y;
  int total   = (M >> 4) * tiles_n;
  if (tile >= total) return;                  // uniform per wave
  int tm = tile / tiles_n, tn = tile % tiles_n;
  int l = threadIdx.x, half = l >> 4, lm = l & 15;

  v8f c; for (int i = 0; i < 8; ++i) c[i] = 0.f;
  const float* arow = A + (size_t)(tm * 16 + lm) * lda + half * 2;
  const float* bcol = B + (size_t)(half * 2) * ldb + tn * 16 + lm;
  for (int k0 = 0; k0 < K; k0 += 4) {
    v2f a, b;
    a.x = arow[k0 + 0];
    a.y = arow[k0 + 1];
    b.x = bcol[(size_t)(k0 + 0) * ldb];
    b.y = bcol[(size_t)(k0 + 1) * ldb];
    c = wmma_f32(a, b, c);
  }
  int col = tn * 16 + lm;
  float bv = bias ? bias[col] : 0.f;
  for (int r = 0; r < 8; ++r) {
    int row = tm * 16 + r + 8 * half;
    float v = c[r] + bv;
    if (relu) v = fmaxf(v, 0.f);
    C[(size_t)row * ldc + col] = v;
  }
}

// ---------------------------------------------------------------------------
// One modern-Hopfield attention sweep for one (batch-tile, head) per wave.
// Online (flash) softmax over chunks of 16 stored patterns.
//   readout==0 : q <- softmax(scale * q K^T) K  (in place), accumulate ||dq||^2
//   readout==1 : out <- softmax(scale * q K^T) V
// ---------------------------------------------------------------------------
template <int DH>
__global__ void hopfield_iter_kernel(float* __restrict__ q,        // [B, hidden]
                                     const float* __restrict__ Kmat,// [N, hidden]
                                     const float* __restrict__ Bmat,// K or V
                                     float* __restrict__ outp,      // [B, hidden]
                                     int Nst, int hidden,
                                     float* __restrict__ diff_acc,
                                     const int* __restrict__ done,
                                     int readout) {
  if (!readout && *done) return;              // uniform across wave
  constexpr int KS = DH / 4;                  // k-steps for logits
  constexpr int DT = DH / 16;                 // 16-wide d tiles
  int bt = blockIdx.x, h = blockIdx.y;
  int l = threadIdx.x, half = l >> 4, lm = l & 15;
  const float* kh = Kmat + h * DH;
  const float* bh = Bmat + h * DH;
  float*       qh = q    + h * DH;

  // q tile in A layout (constant across chunks)
  v2f aq[KS];
  {
    const float* qr = qh + (size_t)(bt * 16 + lm) * hidden + half * 2;
    for (int s = 0; s < KS; ++s) { aq[s].x = qr[s * 4]; aq[s].y = qr[s * 4 + 1]; }
  }

  float mrow[8], lrow[8];
  v8f acc[DT];
  for (int r = 0; r < 8; ++r) { mrow[r] = NEG_INF; lrow[r] = 0.f; }
  for (int t = 0; t < DT; ++t) for (int i = 0; i < 8; ++i) acc[t][i] = 0.f;

  __shared__ float lds_p[256];                // 16x16 staging (1 wave / block)

  for (int nb = 0; nb < (Nst >> 4); ++nb) {
    // logits tile: scale * q @ k_chunk^T
    v8f s; for (int i = 0; i < 8; ++i) s[i] = 0.f;
    {
      const float* kr = kh + (size_t)(nb * 16 + lm) * hidden + half * 2;
      for (int ks = 0; ks < KS; ++ks) {
        v2f b; b.x = kr[ks * 4]; b.y = kr[ks * 4 + 1];
        s = wmma_f32(aq[ks], b, s);
      }
    }
    // online softmax (row stats reduced across each 16-lane half)
    float pr[8];
    for (int r = 0; r < 8; ++r) {
      float v = s[r] * SCALING;
      float tmax = v;
      for (int o = 1; o < 16; o <<= 1) tmax = fmaxf(tmax, __shfl_xor(tmax, o, 32));
      float newm = fmaxf(mrow[r], tmax);
      float corr = __expf(mrow[r] - newm);
      lrow[r] *= corr;
      for (int t = 0; t < DT; ++t) acc[t][r] *= corr;
      float p = __expf(v - newm);
      float rs = p;
      for (int o = 1; o < 16; o <<= 1) rs += __shfl_xor(rs, o, 32);
      lrow[r] += rs;
      mrow[r] = newm;
      pr[r] = p;
    }
    // stage P (C layout) -> LDS -> reload in A layout
    __syncthreads();
    for (int r = 0; r < 8; ++r) lds_p[(r + 8 * half) * 16 + lm] = pr[r];
    __syncthreads();
    // acc += P @ B_chunk
    for (int ks = 0; ks < 4; ++ks) {
      v2f ap;
      ap.x = lds_p[lm * 16 + ks * 4 + half * 2 + 0];
      ap.y = lds_p[lm * 16 + ks * 4 + half * 2 + 1];
      int kr = nb * 16 + ks * 4 + half * 2;
      for (int t = 0; t < DT; ++t) {
        v2f bb;
        bb.x = bh[(size_t)(kr + 0) * hidden + t * 16 + lm];
        bb.y = bh[(size_t)(kr + 1) * hidden + t * 16 + lm];
        acc[t] = wmma_f32(ap, bb, acc[t]);
      }
    }
  }

  if (!readout) {
    float dsum = 0.f;
    for (int t = 0; t < DT; ++t)
      for (int r = 0; r < 8; ++r) {
        int row = bt * 16 + r + 8 * half, col = t * 16 + lm;
        float qn = acc[t][r] / lrow[r];
        float qo = qh[(size_t)row * hidden + col];
        float d = qn - qo;
        dsum += d * d;
        qh[(size_t)row * hidden + col] = qn;
      }
    for (int o = 1; o < 32; o <<= 1) dsum += __shfl_xor(dsum, o, 32);
    if (l == 0) atomicAdd(diff_acc, dsum);
  } else {
    float* oh = outp + h * DH;
    for (int t = 0; t < DT; ++t)
      for (int r = 0; r < 8; ++r) {
        int row = bt * 16 + r + 8 * half, col = t * 16 + lm;
        oh[(size_t)row * hidden + col] = acc[t][r] / lrow[r];
      }
  }
}

__global__ void init_state_kernel(float* diff, int* done) { *diff = 0.f; *done = 0; }

__global__ void conv_check_kernel(float* diff, int* done) {
  if (*diff < EPS2) *done = 1;
  *diff = 0.f;
}

// ---------------------------------------------------------------------------
// Normalize query rows (one block per row).
// ---------------------------------------------------------------------------
__global__ void qnorm_kernel(const float* __restrict__ q, float* __restrict__ qn) {
  int b = blockIdx.x, t = threadIdx.x;
  __shared__ float red[256];
  float s = 0.f;
  for (int c = t; c < DIM; c += 256) { float v = q[b * DIM + c]; s += v * v; }
  red[t] = s; __syncthreads();
  for (int o = 128; o > 0; o >>= 1) { if (t < o) red[t] += red[t + o]; __syncthreads(); }
  float inv = 1.f / fmaxf(sqrtf(red[0]), 1e-8f);
  for (int c = t; c < DIM; c += 256) qn[b * DIM + c] = q[b * DIM + c] * inv;
}

// Inverse norms of memory rows (one wave per row, 8 waves/block).
__global__ void meminv_kernel(const float* __restrict__ mem, float* __restrict__ inv) {
  int wave = threadIdx.x >> 5, lane = threadIdx.x & 31;
  int row = blockIdx.x * 8 + wave;
  if (row >= NMEM) return;
  float s = 0.f;
  for (int c = lane; c < DIM; c += 32) { float v = mem[(size_t)row * DIM + c]; s += v * v; }
  for (int o = 1; o < 32; o <<= 1) s += __shfl_xor(s, o, 32);
  if (lane == 0) inv[row] = 1.f / fmaxf(sqrtf(s), 1e-8f);
}

__global__ void copy_counts_kernel(const float* __restrict__ src, float* __restrict__ dst) {
  int i = blockIdx.x * blockDim.x + threadIdx.x;
  if (i < NMEM) dst[i] = src[i];
}

// ---------------------------------------------------------------------------
// Cosine sims via WMMA + per-chunk top-5.
// Block = 8 waves; wave w handles batch tiles {2w,2w+1}; chunk = 25 mem tiles
// (400 rows). 6250 mem tiles total -> 250 chunks.
// ---------------------------------------------------------------------------
#define MT_PER_CHUNK 25
#define NCHUNKS 250

__device__ __forceinline__ void topk_insert(float v, int idx, float* tv, int* ti) {
  for (int s = 0; s < TOPK; ++s) {
    if (v > tv[s] || (v == tv[s] && idx < ti[s])) {
      for (int z = TOPK - 1; z > s; --z) { tv[z] = tv[z - 1]; ti[z] = ti[z - 1]; }
      tv[s] = v; ti[s] = idx;
      return;
    }
  }
}

__global__ void sims_topk_partial_kernel(const float* __restrict__ qn,
                                         const float* __restrict__ mem,
                                         const float* __restrict__ inv,
                                         float* __restrict__ pv,
                                         int* __restrict__ pi) {
  int chunk = blockIdx.x;
  int wave = threadIdx.x >> 5, l = threadIdx.x & 31;
  int half = l >> 4, lm = l & 15;
  __shared__ float lds[8 * 256];

  for (int bi = 0; bi < 2; ++bi) {
    int bt = wave * 2 + bi;
    float tv[TOPK]; int ti[TOPK];
    for (int s = 0; s < TOPK; ++s) { tv[s] = NEG_INF; ti[s] = 0x7fffffff; }

    for (int mt = 0; mt < MT_PER_CHUNK; ++mt) {
      int jbase = (chunk * MT_PER_CHUNK + mt) * 16;
      v8f c; for (int i = 0; i < 8; ++i) c[i] = 0.f;
      const float* ar = qn  + (size_t)(bt * 16 + lm) * DIM + half * 2;
      const float* br = mem + (size_t)(jbase + lm)   * DIM + half * 2;   // B = mem^T
      for (int k0 = 0; k0 < DIM; k0 += 4) {
        v2f a, b;
        a.x = ar[k0]; a.y = ar[k0 + 1];
        b.x = br[k0]; b.y = br[k0 + 1];
        c = wmma_f32(a, b, c);
      }
      float invn = inv[jbase + lm];                    // per column n = lm
      __syncthreads();
      for (int r = 0; r < 8; ++r)
        lds[wave * 256 + (r + 8 * half) * 16 + lm] = c[r] * invn;
      __syncthreads();
      if (half == 0) {                                 // lanes 0-15 own rows
        for (int n = 0; n < 16; ++n)
          topk_insert(lds[wave * 256 + lm * 16 + n], jbase + n, tv, ti);
      }
    }
    if (half == 0) {
      int row = bt * 16 + lm;
      for (int s = 0; s < TOPK; ++s) {
        pv[(size_t)(chunk * BATCH + row) * TOPK + s] = tv[s];
        pi[(size_t)(chunk * BATCH + row) * TOPK + s] = ti[s];
      }
    }
  }
}

// Merge 250 partial top-5s per row; write scores; bump counts.
__global__ void topk_merge_kernel(const float* __restrict__ pv,
                                  const int* __restrict__ pi,
                                  float* __restrict__ out_scores,
                                  float* __restrict__ counts) {
  int row = blockIdx.x, t = threadIdx.x;          // 64 threads
  __shared__ float sv[64 * TOPK];
  __shared__ int   si[64 * TOPK];
  float tv[TOPK]; int ti[TOPK];
  for (int s = 0; s < TOPK; ++s) { tv[s] = NEG_INF; ti[s] = 0x7fffffff; }
  for (int c = t; c < NCHUNKS; c += 64)
    for (int s = 0; s < TOPK; ++s)
      topk_insert(pv[(size_t)(c * BATCH + row) * TOPK + s],
                  pi[(size_t)(c * BATCH + row) * TOPK + s], tv, ti);
  for (int s = 0; s < TOPK; ++s) { sv[t * TOPK + s] = tv[s]; si[t * TOPK + s] = ti[s]; }
  __syncthreads();
  if (t == 0) {
    float fv[TOPK]; int fi[TOPK];
    for (int s = 0; s < TOPK; ++s) { fv[s] = NEG_INF; fi[s] = 0x7fffffff; }
    for (int e = 0; e < 64 * TOPK; ++e) topk_insert(sv[e], si[e], fv, fi);
    for (int s = 0; s < TOPK; ++s) {
      out_scores[row * TOPK + s] = fv[s];
      atomicAdd(&counts[fi[s]], 1.f);
    }
  }
}

// ---------------------------------------------------------------------------
// Host-side orchestration (all launches on `stream`; graph-capturable).
// ---------------------------------------------------------------------------
static inline void gemm(const float* A, int lda, const float* B, int ldb,
                        float* C, int ldc, const float* bias, int relu,
                        int M, int N, int K, hipStream_t stream) {
  int tiles = (M / 16) * (N / 16);
  dim3 blk(32, 4);
  int grid = (tiles + 3) / 4;
  gemm_wmma_kernel<<<grid, blk, 0, stream>>>(A, lda, B, ldb, C, ldc, bias, relu, M, N, K);
}

extern "C" void kernel_launch(void* const* d_in, const int* in_sizes, int n_in,
                              void* d_out, int out_size, void* d_ws, size_t ws_size,
                              hipStream_t stream) {
  const float* query = (const float*)d_in[0];
  const float* Wq0 = (const float*)d_in[1];
  const float* Wk0 = (const float*)d_in[2];
  const float* Wv0 = (const float*)d_in[3];
  const float* Wo0 = (const float*)d_in[4];
  const float* S0  = (const float*)d_in[5];
  const float* Wq1 = (const float*)d_in[6];
  const float* Wk1 = (const float*)d_in[7];
  const float* Wv1 = (const float*)d_in[8];
  const float* Wo1 = (const float*)d_in[9];
  const float* S1  = (const float*)d_in[10];
  const float* Wc1 = (const float*)d_in[11];
  const float* bc1 = (const float*)d_in[12];
  const float* Wc2 = (const float*)d_in[13];
  const float* bc2 = (const float*)d_in[14];
  const float* mem = (const float*)d_in[15];
  const float* ac  = (const float*)d_in[16];

  float* out = (float*)d_out;
  float* out_scores = out + BATCH * DIM;                       // [256,5]
  float* out_counts = out_scores + BATCH * TOPK;               // [100000]

  float* ws = (float*)d_ws;
  size_t off = 0;
  float* K0   = ws + off; off += (size_t)NSTORED * 256;
  float* V0   = ws + off; off += (size_t)NSTORED * 256;
  float* K1   = ws + off; off += (size_t)NSTORED * 128;
  float* V1   = ws + off; off += (size_t)NSTORED * 128;
  float* Q0   = ws + off; off += (size_t)BATCH * 256;
  float* Q1   = ws + off; off += (size_t)BATCH * 128;
  float* att0 = ws + off; off += (size_t)BATCH * 256;
  float* att1 = ws + off; off += (size_t)BATCH * 128;
  float* cat  = ws + off; off += (size_t)BATCH * 1024;
  float* hbuf = ws + off; off += (size_t)BATCH * DIM;
  float* qn   = ws + off; off += (size_t)BATCH * DIM;
  float* invn = ws + off; off += (size_t)NMEM;
  float* pv   = ws + off; off += (size_t)NCHUNKS * BATCH * TOPK;
  int*   pi   = (int*)(ws + off); off += (size_t)NCHUNKS * BATCH * TOPK;
  float* diff = ws + off; off += 1;
  int*   done = (int*)(ws + off); off += 1;

  // ---- Hopfield layer 0 (hidden=256, dh=32) ----
  gemm(S0, DIM, Wk0, 256, K0, 256, nullptr, 0, NSTORED, 256, DIM, stream);
  gemm(S0, DIM, Wv0, 256, V0, 256, nullptr, 0, NSTORED, 256, DIM, stream);
  gemm(query, DIM, Wq0, 256, Q0, 256, nullptr, 0, BATCH, 256, DIM, stream);
  init_state_kernel<<<1, 1, 0, stream>>>(diff, done);
  for (int it = 0; it < MAX_ITERS; ++it) {
    hopfield_iter_kernel<32><<<dim3(16, NHEADS), 32, 0, stream>>>(
        Q0, K0, K0, nullptr, NSTORED, 256, diff, done, 0);
    conv_check_kernel<<<1, 1, 0, stream>>>(diff, done);
  }
  hopfield_iter_kernel<32><<<dim3(16, NHEADS), 32, 0, stream>>>(
      Q0, K0, V0, att0, NSTORED, 256, diff, done, 1);

  // ---- Hopfield layer 1 (hidden=128, dh=16) ----
  gemm(S1, DIM, Wk1, 128, K1, 128, nullptr, 0, NSTORED, 128, DIM, stream);
  gemm(S1, DIM, Wv1, 128, V1, 128, nullptr, 0, NSTORED, 128, DIM, stream);
  gemm(query, DIM, Wq1, 128, Q1, 128, nullptr, 0, BATCH, 128, DIM, stream);
  init_state_kernel<<<1, 1, 0, stream>>>(diff, done);
  for (int it = 0; it < MAX_ITERS; ++it) {
    hopfield_iter_kernel<16><<<dim3(16, NHEADS), 32, 0, stream>>>(
        Q1, K1, K1, nullptr, NSTORED, 128, diff, done, 0);
    conv_check_kernel<<<1, 1, 0, stream>>>(diff, done);
  }
  hopfield_iter_kernel<16><<<dim3(16, NHEADS), 32, 0, stream>>>(
      Q1, K1, V1, att1, NSTORED, 128, diff, done, 1);

  // ---- output projections into concat buffer, then consolidation MLP ----
  gemm(att0, 256, Wo0, DIM, cat,       1024, nullptr, 0, BATCH, DIM, 256, stream);
  gemm(att1, 128, Wo1, DIM, cat + DIM, 1024, nullptr, 0, BATCH, DIM, 128, stream);
  gemm(cat, 1024, Wc1, DIM, hbuf, DIM, bc1, 1, BATCH, DIM, 1024, stream);  // ReLU
  gemm(hbuf, DIM, Wc2, DIM, out,  DIM, bc2, 0, BATCH, DIM, DIM, stream);

  // ---- cosine similarity + top-k + access counts ----
  qnorm_kernel<<<BATCH, 256, 0, stream>>>(query, qn);
  meminv_kernel<<<NMEM / 8, 256, 0, stream>>>(mem, invn);
  copy_counts_kernel<<<(NMEM + 255) / 256, 256, 0, stream>>>(ac, out_counts);
  sims_topk_partial_kernel<<<NCHUNKS, 256, 0, stream>>>(qn, mem, invn, pv, pi);
  topk_merge_kernel<<<BATCH, 64, 0, stream>>>(pv, pi, out_scores, out_counts);
}